// AttnPool3D_5514738008256
// MI455X (gfx1250) — compile-verified
//
#include <hip/hip_runtime.h>

// Problem constants (fixed by the reference): B=2, C=128, N=48*96*96
#define C_DIM    128
#define N_DIM    442368
#define TILE_N   256
#define NTILES   1728            // N_DIM / TILE_N (exact)
#define ROW_F    268             // padded LDS row stride in floats (16B aligned, bank-aware)
#define NTHREADS 256
#define MASK_BIAS 2.0f

typedef float v2f __attribute__((ext_vector_type(2)));
typedef float v8f __attribute__((ext_vector_type(8)));
typedef int   v4i __attribute__((ext_vector_type(4)));

// LDS layout (floats), dynamic shared
#define OFF_TILE   0
#define OFF_W      (C_DIM * ROW_F)          // 34304
#define OFF_LOGITS (OFF_W + C_DIM)          // 34432
#define OFF_E      (OFF_LOGITS + TILE_N)    // 34688
#define OFF_RED    (OFF_E + TILE_N)         // 34944
#define SMEM_FLOATS (OFF_RED + 64)          // 35008 floats = 140032 B

__device__ __forceinline__ void async_load_b128(const float* g, float* l) {
#if __has_builtin(__builtin_amdgcn_global_load_async_to_lds_b128)
  typedef __attribute__((address_space(1))) v4i GV;   // global int4*
  typedef __attribute__((address_space(3))) v4i LV;   // LDS int4*
  // generic->flat value: global flat VA == AS1 VA; LDS flat low 32 bits == AS3 offset (ISA 10.2)
  __builtin_amdgcn_global_load_async_to_lds_b128(
      (GV*)(unsigned long long)(size_t)g,
      (LV*)(unsigned)(size_t)l, 0, 0);
#else
  unsigned      la = (unsigned)(size_t)l;
  unsigned long long ga = (unsigned long long)(size_t)g;
  asm volatile("global_load_async_to_lds_b128 %0, %1, off" :: "v"(la), "v"(ga) : "memory");
#endif
}

__device__ __forceinline__ void wait_async0() {
#if __has_builtin(__builtin_amdgcn_s_wait_asynccnt)
  __builtin_amdgcn_s_wait_asynccnt(0);
#else
  asm volatile("s_wait_asynccnt 0x0" ::: "memory");
#endif
}

__device__ __forceinline__ float wave_max(float v) {
  for (int o = 16; o > 0; o >>= 1) v = fmaxf(v, __shfl_xor(v, o, 32));
  return v;
}
__device__ __forceinline__ float wave_sum(float v) {
  for (int o = 16; o > 0; o >>= 1) v += __shfl_xor(v, o, 32);
  return v;
}

// One block = one (batch, 256-position tile). Emits per-tile (max m, sumexp s, p[128]).
__global__ void attnpool_tile_kernel(const float* __restrict__ feat,
                                     const float* __restrict__ mask,
                                     const float* __restrict__ w_attn,
                                     float* __restrict__ ws_p,
                                     float* __restrict__ ws_m,
                                     float* __restrict__ ws_s) {
  extern __shared__ float smem[];
  float* tile     = smem + OFF_TILE;    // [128][ROW_F]
  float* w_s      = smem + OFF_W;       // [128]
  float* logits_s = smem + OFF_LOGITS;  // [256]
  float* e_s      = smem + OFF_E;       // [256]
  float* red_s    = smem + OFF_RED;     // [64]

  const int t    = threadIdx.x;
  const int lane = t & 31;
  const int half = lane >> 4;           // 0: lanes 0-15, 1: lanes 16-31
  const int lL   = lane & 15;
  const int wid  = t >> 5;              // 8 waves
  const int tileIdx = blockIdx.x;       // 0..NTILES-1
  const int b       = blockIdx.y;       // 0..1
  const int n0      = tileIdx * TILE_N;
  const int blk     = b * NTILES + tileIdx;

  // ---- Stage feat tile (128 c x 256 n, fp32) into LDS via async b128 DMA ----
  const float* gbase = feat + (size_t)b * C_DIM * N_DIM + n0;
  for (int i = 0; i < 32; ++i) {              // 8192 b128 chunks / 256 threads
    int chunk = i * NTHREADS + t;
    int c = chunk >> 6;                       // 64 x b128 per row
    int k = chunk & 63;
    async_load_b128(gbase + (size_t)c * N_DIM + (k << 2),
                    tile + c * ROW_F + (k << 2));
  }
  if (t < C_DIM) w_s[t] = w_attn[t];
  wait_async0();
  __syncthreads();

  // ---- Phase A: logits[n] = sum_c tile[c][n] * w[c] via V_WMMA_F32_16X16X4_F32 ----
  // A: M=n (16), K=c (4). lane(half,lL): M=lL, v0->K=2*half, v1->K=2*half+1.
  // B[k][j] = w[c0+k] for all j  =>  D[m][j] = logit(n=nb+m) in every column j.
  for (int g = 0; g < 2; ++g) {
    const int nb = (wid * 2 + g) * 16;
    v8f acc = {0.f, 0.f, 0.f, 0.f, 0.f, 0.f, 0.f, 0.f};
    for (int c0 = 0; c0 < C_DIM; c0 += 4) {
      const int ca = c0 + 2 * half;
      v2f a;  a.x = tile[ca * ROW_F + nb + lL];
              a.y = tile[(ca + 1) * ROW_F + nb + lL];
      v2f bm; bm.x = w_s[ca];
              bm.y = w_s[ca + 1];
      acc = __builtin_amdgcn_wmma_f32_16x16x4_f32(false, a, false, bm,
                                                  (short)0, acc, false, false);
    }
    if (lL == 0) {  // lane0 -> rows 0..7, lane16 -> rows 8..15 (C/D VGPR layout)
      for (int r = 0; r < 8; ++r) logits_s[nb + half * 8 + r] = acc[r];
    }
  }
  __syncthreads();

  // ---- mask bias + block softmax stats (split-softmax) ----
  float mv = mask[(size_t)b * N_DIM + n0 + t];
  float l  = logits_s[t] + MASK_BIAS * fminf(fmaxf(mv, 0.f), 1.f);

  float m = wave_max(l);
  if (lane == 0) red_s[wid] = m;
  __syncthreads();
  if (t == 0) {
    float mm = red_s[0];
    for (int i = 1; i < 8; ++i) mm = fmaxf(mm, red_s[i]);
    red_s[16] = mm;
  }
  __syncthreads();
  const float M = red_s[16];

  float e = __expf(l - M);
  e_s[t] = e;
  float s = wave_sum(e);
  if (lane == 0) red_s[8 + wid] = s;
  __syncthreads();
  if (t == 0) {
    float ss = 0.f;
    for (int i = 0; i < 8; ++i) ss += red_s[8 + i];
    ws_m[blk] = M;
    ws_s[blk] = ss;
  }
  __syncthreads();   // e_s visible to all waves

  // ---- Phase D: p[c] = sum_n tile[c][n] * e[n] via WMMA (A: M=c, K=n) ----
  {
    const int cg = wid * 16;
    v8f acc = {0.f, 0.f, 0.f, 0.f, 0.f, 0.f, 0.f, 0.f};
    for (int nn = 0; nn < TILE_N; nn += 4) {
      const int na = nn + 2 * half;
      v2f a;  a.x = tile[(cg + lL) * ROW_F + na];
              a.y = tile[(cg + lL) * ROW_F + na + 1];
      v2f bm; bm.x = e_s[na];
              bm.y = e_s[na + 1];
      acc = __builtin_amdgcn_wmma_f32_16x16x4_f32(false, a, false, bm,
                                                  (short)0, acc, false, false);
    }
    if (lL == 0) {
      for (int r = 0; r < 8; ++r)
        ws_p[(size_t)blk * C_DIM + cg + half * 8 + r] = acc[r];
    }
  }
}

// One block per batch: merge 1728 tile-partials into out[b][c].
__global__ void attnpool_finalize_kernel(const float* __restrict__ ws_p,
                                         const float* __restrict__ ws_m,
                                         const float* __restrict__ ws_s,
                                         float* __restrict__ out) {
  __shared__ float scale[NTILES];
  __shared__ float red[72];
  const int b = blockIdx.x;
  const int t = threadIdx.x;           // 256 threads
  const int lane = t & 31, wid = t >> 5;
  const int base = b * NTILES;

  float m = -3.0e38f;
  for (int i = t; i < NTILES; i += NTHREADS) m = fmaxf(m, ws_m[base + i]);
  m = wave_max(m);
  if (lane == 0) red[wid] = m;
  __syncthreads();
  if (t == 0) {
    float mm = red[0];
    for (int i = 1; i < 8; ++i) mm = fmaxf(mm, red[i]);
    red[64] = mm;
  }
  __syncthreads();
  const float M = red[64];

  float s = 0.f;
  for (int i = t; i < NTILES; i += NTHREADS) {
    float sc = __expf(ws_m[base + i] - M);
    scale[i] = sc;
    s += ws_s[base + i] * sc;
  }
  s = wave_sum(s);
  if (lane == 0) red[8 + wid] = s;
  __syncthreads();
  if (t == 0) {
    float ss = 0.f;
    for (int i = 0; i < 8; ++i) ss += red[8 + i];
    red[65] = ss;
  }
  __syncthreads();
  const float S = red[65];

  if (t < C_DIM) {
    float acc = 0.f;
    for (int i = 0; i < NTILES; ++i)
      acc += ws_p[((size_t)base + i) * C_DIM + t] * scale[i];
    out[b * C_DIM + t] = acc / S;
  }
}

extern "C" void kernel_launch(void* const* d_in, const int* in_sizes, int n_in,
                              void* d_out, int out_size, void* d_ws, size_t ws_size,
                              hipStream_t stream) {
  const float* feat   = (const float*)d_in[0];
  const float* mask   = (const float*)d_in[1];
  const float* w_attn = (const float*)d_in[2];
  float* out = (float*)d_out;

  float* ws   = (float*)d_ws;
  float* ws_p = ws;                                   // [2*NTILES][128]
  float* ws_m = ws + (size_t)2 * NTILES * C_DIM;      // [2*NTILES]
  float* ws_s = ws_m + (size_t)2 * NTILES;            // [2*NTILES]

  dim3 grid(NTILES, 2);
  size_t shmem = (size_t)SMEM_FLOATS * sizeof(float); // 140032 B < 160 KB (2 blocks/WGP)
  attnpool_tile_kernel<<<grid, NTHREADS, shmem, stream>>>(feat, mask, w_attn,
                                                          ws_p, ws_m, ws_s);
  attnpool_finalize_kernel<<<2, NTHREADS, 0, stream>>>(ws_p, ws_m, ws_s, out);
}